// Texture_53283364274864
// MI455X (gfx1250) — compile-verified
//
#include <hip/hip_runtime.h>

typedef float v4f __attribute__((ext_vector_type(4)));
typedef int   v4i __attribute__((vector_size(16)));   // matches builtin's pointee

namespace {

constexpr int THREADS       = 256;                      // 8 waves (wave32)
constexpr int GROUPS        = 8;                        // 4-pixel groups per thread
constexpr int PIX_PER_BLOCK = THREADS * 4 * GROUPS;     // 8192
constexpr int TOTAL_PIX     = 16 * 1024 * 1024;         // B*HO*WO

// Bilinear sample with zero padding (align_corners=false), matches reference.
// Works for both global-memory textures and the LDS-staged level-4 texture
// (after inlining, InferAddressSpaces specializes the loads to ds_load).
__device__ __forceinline__ float fetch(const float* __restrict__ t, int N,
                                       int xi, int yi) {
    float v = 0.0f;
    if (((unsigned)xi < (unsigned)N) & ((unsigned)yi < (unsigned)N))
        v = t[yi * N + xi];
    return v;
}

__device__ __forceinline__ float sample(const float* __restrict__ t, const int N,
                                        float gx, float gy) {
    const float h  = 0.5f * (float)N;
    const float ix = fmaf(gx, h, h - 0.5f);   // ((gx+1)*N - 1) * 0.5
    const float iy = fmaf(gy, h, h - 0.5f);
    const float x0f = floorf(ix);
    const float y0f = floorf(iy);
    const float wx1 = ix - x0f;
    const float wy1 = iy - y0f;
    const float wx0 = 1.0f - wx1;
    const float wy0 = 1.0f - wy1;
    const int x0 = (int)x0f;
    const int y0 = (int)y0f;
    const float v00 = fetch(t, N, x0,     y0);
    const float v10 = fetch(t, N, x0 + 1, y0);
    const float v01 = fetch(t, N, x0,     y0 + 1);
    const float v11 = fetch(t, N, x0 + 1, y0 + 1);
    return wx0 * wy0 * v00 + wx1 * wy0 * v10 + wx0 * wy1 * v01 + wx1 * wy1 * v11;
}

__global__ void __launch_bounds__(THREADS)
texture_pyramid_kernel(const float* __restrict__ x,
                       const float* __restrict__ l1,
                       const float* __restrict__ l2,
                       const float* __restrict__ l3,
                       const float* __restrict__ l4,
                       float* __restrict__ out) {
    __shared__ float l4s[128 * 128];            // 64 KB: whole level-4 texture

    const int tid = threadIdx.x;

    // ---- Stage level-4 into LDS via async global->LDS (ASYNCcnt path) ----
#if __has_builtin(__builtin_amdgcn_global_load_async_to_lds_b128) && \
    __has_builtin(__builtin_amdgcn_s_wait_asynccnt)
    {
        typedef __attribute__((address_space(1))) v4i* gvp_t;   // global
        typedef __attribute__((address_space(3))) v4i* lvp_t;   // LDS
        gvp_t gsrc = (gvp_t)l4;       // C-style cast drops const
        lvp_t ldst = (lvp_t)l4s;
        #pragma unroll
        for (int j = 0; j < 16; ++j) {
            const int idx = j * THREADS + tid;          // 16 B elements
            __builtin_amdgcn_global_load_async_to_lds_b128(gsrc + idx, ldst + idx,
                                                           0, 0);
        }
        __builtin_amdgcn_s_wait_asynccnt(0);
    }
#else
    #pragma unroll
    for (int j = 0; j < 16; ++j) {
        ((v4f*)l4s)[j * THREADS + tid] = ((const v4f*)l4)[j * THREADS + tid];
    }
#endif
    __syncthreads();

    const long blockBase = (long)blockIdx.x * PIX_PER_BLOCK;

    for (int g = 0; g < GROUPS; ++g) {
        const long p  = blockBase + (long)(g * THREADS + tid) * 4; // first of 4 px
        const int  b  = (int)(p >> 20);                            // HO*WO = 2^20
        const int  hw = (int)(p & 0xFFFFF);

        // 4 pixels' (x,y) coords: 32 contiguous bytes, read-once -> NT loads.
        const v4f* xp  = (const v4f*)(x + p * 2);
        const v4f  c01 = __builtin_nontemporal_load(xp);
        const v4f  c23 = __builtin_nontemporal_load(xp + 1);

        float gx[4], gy[4];
        gx[0] = fmaf(c01.x, 2.0f, -1.0f);  gy[0] = fmaf(c01.y, 2.0f, -1.0f);
        gx[1] = fmaf(c01.z, 2.0f, -1.0f);  gy[1] = fmaf(c01.w, 2.0f, -1.0f);
        gx[2] = fmaf(c23.x, 2.0f, -1.0f);  gy[2] = fmaf(c23.y, 2.0f, -1.0f);
        gx[3] = fmaf(c23.z, 2.0f, -1.0f);  gy[3] = fmaf(c23.w, 2.0f, -1.0f);

        v4f yv;
        #pragma unroll
        for (int i = 0; i < 4; ++i) {
            float y = sample(l1, 1024, gx[i], gy[i]);   // global (L2-resident)
            y      += sample(l2,  512, gx[i], gy[i]);   // global (L2-resident)
            y      += sample(l3,  256, gx[i], gy[i]);   // global (L2-resident)
            y      += sample(l4s, 128, gx[i], gy[i]);   // LDS
            yv[i] = y;
        }

        // y1 == y2 == y3 in the reference: write once per channel, NT stores
        // (output is 201 MB write-once; keep textures resident in L2).
        float* o = out + (((long)b * 3) << 20) + hw;
        __builtin_nontemporal_store(yv, (v4f*)o);
        __builtin_nontemporal_store(yv, (v4f*)(o + (1 << 20)));
        __builtin_nontemporal_store(yv, (v4f*)(o + (2 << 20)));
    }
}

} // namespace

extern "C" void kernel_launch(void* const* d_in, const int* in_sizes, int n_in,
                              void* d_out, int out_size, void* d_ws, size_t ws_size,
                              hipStream_t stream) {
    // setup_inputs order: x, p1_l1..p1_l4, p2_l1..p2_l4, p3_l1..p3_l4.
    // Reference only ever uses p1_* (pyramid1 sampled three times).
    const float* x  = (const float*)d_in[0];
    const float* l1 = (const float*)d_in[1];   // 1024x1024
    const float* l2 = (const float*)d_in[2];   //  512x512
    const float* l3 = (const float*)d_in[3];   //  256x256
    const float* l4 = (const float*)d_in[4];   //  128x128
    float* out = (float*)d_out;                // [16, 3, 1024, 1024] f32

    const int blocks = TOTAL_PIX / PIX_PER_BLOCK;   // 2048
    texture_pyramid_kernel<<<blocks, THREADS, 0, stream>>>(x, l1, l2, l3, l4, out);
}